// FaceOffsetSymmetricReg_41970420417779
// MI455X (gfx1250) — compile-verified
//
#include <hip/hip_runtime.h>
#include <hip/hip_bf16.h>
#include <stdint.h>

// ---------------------------------------------------------------------------
// FaceOffsetSymmetricReg on MI455X (gfx1250)
//
// Bandwidth-bound gather problem (~82 MB total traffic, ~0.1 GFLOP).
//   1) Prep kernels build a per-face gather table (b-independent), hot in L2.
//   2) Main kernel: one workgroup per batch b. Stream face_offset[b]
//      (F*3 floats ~= 60 KB) into LDS with CDNA5 async global->LDS loads
//      (B128 bulk + B32 head/tail, phase-shifted LDS base so both sides are
//      16B aligned), s_wait_asynccnt 0, barrier, then all-F LDS gathers.
// ---------------------------------------------------------------------------

typedef int v4i __attribute__((__vector_size__(4 * sizeof(int))));

typedef __attribute__((address_space(1))) int as1_int;
typedef __attribute__((address_space(3))) int as3_int;
typedef __attribute__((address_space(1))) v4i as1_v4i;
typedef __attribute__((address_space(3))) v4i as3_v4i;

__device__ __forceinline__ void async_g2l_b32(const void* g, void* l) {
    __builtin_amdgcn_global_load_async_to_lds_b32((as1_int*)g, (as3_int*)l, 0, 0);
}
__device__ __forceinline__ void async_g2l_b128(const void* g, void* l) {
    __builtin_amdgcn_global_load_async_to_lds_b128((as1_v4i*)g, (as3_v4i*)l, 0, 0);
}
__device__ __forceinline__ void wait_async_zero() {
    asm volatile("s_wait_asynccnt 0" ::: "memory");
}

// ---- prep kernel 1: inv[v] = -1 --------------------------------------------
__global__ void k_init_inv(int* __restrict__ inv, int V) {
    int v = blockIdx.x * blockDim.x + threadIdx.x;
    if (v < V) inv[v] = -1;
}

// ---- prep kernel 2: inv[fvi[f]] = max(f) -----------------------------------
__global__ void k_scatter_inv(const int* __restrict__ fvi,
                              int* __restrict__ inv, int F) {
    int f = blockIdx.x * blockDim.x + threadIdx.x;
    if (f < F) atomicMax(&inv[fvi[f]], f);
}

// ---- prep kernel 3: per-face gather table ----------------------------------
__global__ void k_build_table(const int*   __restrict__ fvi,
                              const int*   __restrict__ cfaces,   // (V,3)
                              const float* __restrict__ bc,       // (V,3)
                              const int*   __restrict__ inv,
                              int4*   __restrict__ tidx,
                              float4* __restrict__ tbc, int F) {
    int f = blockIdx.x * blockDim.x + threadIdx.x;
    if (f >= F) return;
    int v  = fvi[f];
    int w  = inv[v];                 // always >= 0 (v is in image of fvi)
    int g0 = inv[cfaces[3 * v + 0]];
    int g1 = inv[cfaces[3 * v + 1]];
    int g2 = inv[cfaces[3 * v + 2]];
    tidx[f] = make_int4(w, g0, g1, g2);
    tbc[f]  = make_float4(bc[3 * v + 0], bc[3 * v + 1], bc[3 * v + 2], 0.0f);
}

// ---- main kernel: one block per batch element ------------------------------
__global__ void __launch_bounds__(512)
k_face_offset_loss(const float* __restrict__ face_offset,  // (B,F,3)
                   const int4*  __restrict__ tidx,         // (F)
                   const float4*__restrict__ tbc,          // (F)
                   float* __restrict__ out,                // (B,F)
                   int F) {
    extern __shared__ float smem[];                        // F*3 + 4 floats
    const int F3 = F * 3;
    const int b  = blockIdx.x;
    const float* src = face_offset + (size_t)b * F3;

    // Phase-shift LDS base so LDS and global offsets are congruent mod 16B;
    // then the bulk of the copy can use 16B-per-lane async B128 transfers.
    const int r  = (int)(((size_t)b * (size_t)F3) & 3);  // src float misalign
    float* sm    = smem + r;
    const int i0 = (4 - r) & 3;                          // first 16B-aligned f32
    const int nv = (F3 - i0) >> 2;                       // # of float4 chunks
    const int tail0 = i0 + (nv << 2);

    for (int i = threadIdx.x; i < nv; i += blockDim.x) {
        async_g2l_b128(src + i0 + 4 * i, sm + i0 + 4 * i);
    }
    if ((int)threadIdx.x < i0) {                         // head: 0..3 floats
        async_g2l_b32(src + threadIdx.x, sm + threadIdx.x);
    }
    if ((int)threadIdx.x < F3 - tail0) {                 // tail: 0..3 floats
        async_g2l_b32(src + tail0 + threadIdx.x, sm + tail0 + threadIdx.x);
    }
    wait_async_zero();
    __syncthreads();

    for (int f = threadIdx.x; f < F; f += blockDim.x) {
        int4   id = tidx[f];
        float4 wv = tbc[f];

        const float a0 = sm[id.x * 3 + 0];
        const float a1 = sm[id.x * 3 + 1];
        const float a2 = sm[id.x * 3 + 2];

        // branchless masked gathers for the three closest-face contributions
        const float m0 = (id.y >= 0) ? wv.x : 0.0f;
        const float m1 = (id.z >= 0) ? wv.y : 0.0f;
        const float m2 = (id.w >= 0) ? wv.z : 0.0f;
        const int   r0 = (id.y >= 0) ? id.y : 0;
        const int   r1 = (id.z >= 0) ? id.z : 0;
        const int   r2 = (id.w >= 0) ? id.w : 0;

        float s0 = m0 * sm[r0 * 3 + 0] + m1 * sm[r1 * 3 + 0] + m2 * sm[r2 * 3 + 0];
        float s1 = m0 * sm[r0 * 3 + 1] + m1 * sm[r1 * 3 + 1] + m2 * sm[r2 * 3 + 1];
        float s2 = m0 * sm[r0 * 3 + 2] + m1 * sm[r1 * 3 + 2] + m2 * sm[r2 * 3 + 2];

        out[(size_t)b * F + f] = fabsf(a0 + s0) + fabsf(a1 - s1) + fabsf(a2 - s2);
    }
}

// ---------------------------------------------------------------------------
extern "C" void kernel_launch(void* const* d_in, const int* in_sizes, int n_in,
                              void* d_out, int out_size, void* d_ws, size_t ws_size,
                              hipStream_t stream) {
    const float* face_offset = (const float*)d_in[0];   // (B,F,3) f32
    const float* bc          = (const float*)d_in[1];   // (V,3)   f32
    const int*   fvi         = (const int*)  d_in[2];   // (F)
    const int*   cfaces      = (const int*)  d_in[3];   // (V,3)
    float*       out         = (float*)d_out;           // (B,F)

    const int F = in_sizes[2];
    const int V = in_sizes[1] / 3;
    const int B = (int)((long long)in_sizes[0] / (3LL * F));

    // workspace layout
    char*   ws   = (char*)d_ws;
    int*    inv  = (int*)ws;
    size_t  off  = ((size_t)V * sizeof(int) + 15) & ~(size_t)15;
    int4*   tidx = (int4*)(ws + off);
    float4* tbc  = (float4*)(ws + off + (size_t)F * sizeof(int4));

    const int T = 256;
    k_init_inv   <<<(V + T - 1) / T, T, 0, stream>>>(inv, V);
    k_scatter_inv<<<(F + T - 1) / T, T, 0, stream>>>(fvi, inv, F);
    k_build_table<<<(F + T - 1) / T, T, 0, stream>>>(fvi, cfaces, bc, inv, tidx, tbc, F);

    const size_t smem_bytes = ((size_t)F * 3 + 4) * sizeof(float);  // ~60.3 KB
    k_face_offset_loss<<<B, 512, smem_bytes, stream>>>(face_offset, tidx, tbc, out, F);
}